// DFAProb_74079595921830
// MI455X (gfx1250) — compile-verified
//
#include <hip/hip_runtime.h>
#include <stdint.h>

#define EPS    1e-7f
#define NEGINF (-100000.0f)
#define LOG2E  1.4426950408889634f   // 1/ln(2)
#define LN2    0.6931471805599453f

// Raw hardware transcendentals: v_exp_f32 / v_log_f32 are base-2 ops.
#define EXP2F(x) __builtin_amdgcn_exp2f(x)
#define LOG2F(x) __builtin_amdgcn_logf(x)

typedef float        v4f __attribute__((ext_vector_type(4)));
typedef unsigned int v4u __attribute__((ext_vector_type(4)));
typedef int          v4i __attribute__((ext_vector_type(4)));
typedef int          v8i __attribute__((ext_vector_type(8)));

#define RPB  256   // batch rows per block (16 KB tile of log_s0)
#define RPAD 17    // padded LDS row stride in dwords (16 data + 1 pad -> bank-conflict-free)

__global__ __launch_bounds__(RPB) void dfa_fwd(
    const float* __restrict__ log_s0,
    const float* __restrict__ p0, const float* __restrict__ p1,
    const float* __restrict__ p2, const float* __restrict__ p3,
    const float* __restrict__ p4, const float* __restrict__ p5,
    const float* __restrict__ p6, const float* __restrict__ p7,
    float* __restrict__ out_ln, float* __restrict__ out_ns,
    float* __restrict__ out_acc, int B)
{
    __shared__ float tile[RPB * RPAD];
    const int t  = (int)threadIdx.x;
    const int b0 = (int)blockIdx.x * RPB;
    const int b  = b0 + t;
    const int rem = B - b0;
    const unsigned rows = (unsigned)(rem < RPB ? rem : RPB);

    // ---- Stage this block's log_s0 tile (rows x 16 f32) into LDS via TDM ----
#if defined(__gfx1250__) && __has_builtin(__builtin_amdgcn_tensor_load_to_lds)
    if (t == 0) {  // TDM op ignores EXEC; one issue per block (wave 0 only enters)
        unsigned long long ga = (unsigned long long)(uintptr_t)(log_s0 + (size_t)b0 * 16);
        unsigned ldsoff = (unsigned)(uintptr_t)&tile[0];  // low 32 bits = LDS offset

        // D# group 0: count=1 | lds_addr | global_addr[56:0] | type=2
        v4u g0;
        g0.x = 1u;
        g0.y = ldsoff;
        g0.z = (unsigned)(ga & 0xffffffffull);
        g0.w = (unsigned)((ga >> 32) & 0x01ffffffu) | (2u << 30);

        // D# group 1: data_size=4B, pad 1 dword every 16 dwords (row stride 68B),
        // tensor = rows x 16, tile = rows x 16, dim0 stride = 16 elements.
        v8i g1;
        g1[0] = (int)((2u << 16)       // data_size: 4 bytes
                    | (1u << 20)       // pad_enable
                    | (3u << 22)       // pad_interval: every 16 dwords
                    | (0u << 25));     // pad_amount: 1 dword
        g1[1] = (int)(16u << 16);                      // tensor_dim0 = 16 (low16)
        g1[2] = (int)((rows & 0xffffu) << 16);         // dim0 hi=0 | tensor_dim1 lo16
        g1[3] = (int)((rows >> 16) | (16u << 16));     // tensor_dim1 hi | tile_dim0 = 16
        g1[4] = (int)(rows & 0xffffu);                 // tile_dim1 = rows, tile_dim2 = 0
        g1[5] = 16;                                    // tensor_dim0_stride = 16
        g1[6] = 0;
        g1[7] = 0;
        v4i g2 = {0, 0, 0, 0};
        v4i g3 = {0, 0, 0, 0};
#if defined(__clang_major__) && (__clang_major__ >= 23)
        v8i g4 = {0, 0, 0, 0, 0, 0, 0, 0};
        __builtin_amdgcn_tensor_load_to_lds(g0, g1, g2, g3, g4, 0);
#else
        __builtin_amdgcn_tensor_load_to_lds(g0, g1, g2, g3, 0);
#endif
#if __has_builtin(__builtin_amdgcn_s_wait_tensorcnt)
        __builtin_amdgcn_s_wait_tensorcnt(0);
#endif
    }
    __syncthreads();
#else
    for (int i = t; i < (int)rows * 16; i += RPB)
        tile[(i >> 4) * RPAD + (i & 15)] = log_s0[(size_t)b0 * 16 + i];
    __syncthreads();
#endif

    if (b >= B) return;

    const float lo = EPS, hi = 1.0f - EPS;

    // ---- props: streamed once -> non-temporal loads ----
    float pr[8];
    pr[0] = __builtin_nontemporal_load(p0 + b);
    pr[1] = __builtin_nontemporal_load(p1 + b);
    pr[2] = __builtin_nontemporal_load(p2 + b);
    pr[3] = __builtin_nontemporal_load(p3 + b);
    pr[4] = __builtin_nontemporal_load(p4 + b);
    pr[5] = __builtin_nontemporal_load(p5 + b);
    pr[6] = __builtin_nontemporal_load(p6 + b);
    pr[7] = __builtin_nontemporal_load(p7 + b);

    // All log-domain state carried in log2 units: every exp/log below is a
    // single hardware v_exp_f32 / v_log_f32 (base-2), no libm fixup code.
    float lw[8], lnw[8];
#pragma unroll
    for (int i = 0; i < 8; ++i) {
        float pc = fminf(fmaxf(pr[i], lo), hi);
        lw[i]  = LOG2F(pc);           // log2(w_p)
        lnw[i] = LOG2F(1.0f - pc);    // log2(1 - w_p) == log1mexp in log2 units
    }

    // ---- previous state distribution (padded LDS, conflict-free reads), -> log2
    float prev[16];
#pragma unroll
    for (int j = 0; j < 16; ++j)
        prev[j] = fminf(fmaxf(tile[t * RPAD + j], NEGINF), -EPS) * LOG2E;

    // ---- per-target-state 3-term logsumexp (edges of the baked-in DFA):
    //   self loop (~p),  (s-1)->s (p&q),  (s-2)->s (p&~q)
    float unn[16];
#pragma unroll
    for (int s = 0; s < 16; ++s) {
        const int s1 = (s + 15) & 15;
        const int s2 = (s + 14) & 15;
        float ta = prev[s]  + lnw[s & 7];
        float tb = prev[s1] + lw[s1 & 7] + lw[(s1 + 1) & 7];
        float tc = prev[s2] + lw[s2 & 7] + lnw[(s2 + 1) & 7];
        float m  = fmaxf(fmaxf(ta, tb), tc);
        unn[s] = m + LOG2F(EXP2F(ta - m) + EXP2F(tb - m) + EXP2F(tc - m));
    }

    // ---- normalization (reuse exp2(u - m) for both denom and next_state) ----
    float m2 = unn[0];
#pragma unroll
    for (int j = 1; j < 16; ++j) m2 = fmaxf(m2, unn[j]);
    float e[16], sum = 0.0f;
#pragma unroll
    for (int j = 0; j < 16; ++j) { e[j] = EXP2F(unn[j] - m2); sum += e[j]; }
    const float denom = m2 + LOG2F(sum);
    const float invS  = 1.0f / sum;

    float ln[16], ns[16];
#pragma unroll
    for (int j = 0; j < 16; ++j) {
        // back to natural-log units only for the output tensor
        ln[j] = fminf(fmaxf((unn[j] - denom) * LN2, NEGINF), -EPS);
        ns[j] = fminf(fmaxf(e[j] * invS, lo), hi);   // base-independent
    }

    float acc_a = 0.0f, acc_n = 0.0f;
#pragma unroll
    for (int j = 0; j < 8;  ++j) acc_a += ns[j];
#pragma unroll
    for (int j = 8; j < 16; ++j) acc_n += ns[j];
    const float acc_b = 1.0f - acc_n;
    const float acc   = acc_a + acc_b - acc_a * acc_b;

    // ---- streamed outputs -> non-temporal b128 stores ----
    const size_t ro = (size_t)b * 16;
#pragma unroll
    for (int k = 0; k < 4; ++k) {
        v4f vln = { ln[4*k], ln[4*k+1], ln[4*k+2], ln[4*k+3] };
        v4f vns = { ns[4*k], ns[4*k+1], ns[4*k+2], ns[4*k+3] };
        __builtin_nontemporal_store(vln, (v4f*)(out_ln + ro) + k);
        __builtin_nontemporal_store(vns, (v4f*)(out_ns + ro) + k);
    }
    __builtin_nontemporal_store(acc, out_acc + b);
}

extern "C" void kernel_launch(void* const* d_in, const int* in_sizes, int n_in,
                              void* d_out, int out_size, void* d_ws, size_t ws_size,
                              hipStream_t stream) {
    const int B = in_sizes[0] / 16;          // log_s0 is [B, 16]
    const float* log_s0 = (const float*)d_in[0];
    // d_in[1] = s0 (unused by the reference computation)
    const float* p0 = (const float*)d_in[2];
    const float* p1 = (const float*)d_in[3];
    const float* p2 = (const float*)d_in[4];
    const float* p3 = (const float*)d_in[5];
    const float* p4 = (const float*)d_in[6];
    const float* p5 = (const float*)d_in[7];
    const float* p6 = (const float*)d_in[8];
    const float* p7 = (const float*)d_in[9];

    float* out      = (float*)d_out;
    float* out_ln   = out;                          // log_next   [B,16]
    float* out_ns   = out + (size_t)B * 16;         // next_state [B,16]
    float* out_acc  = out + (size_t)B * 32;         // accepted   [B]

    const int blocks = (B + RPB - 1) / RPB;
    dfa_fwd<<<blocks, RPB, 0, stream>>>(log_s0, p0, p1, p2, p3, p4, p5, p6, p7,
                                        out_ln, out_ns, out_acc, B);
}